// HashEmbedding_46136538693901
// MI455X (gfx1250) — compile-verified
//
#include <hip/hip_runtime.h>
#include <hip/hip_bf16.h>
#include <stdint.h>

// HashEmbedding forward: out[r, :] = embedding[hashed_ids[r], :]
//   rows = 8*4096 = 32768, D = 1024 f32 (4 KiB per row), out = 128 MiB.
// Pure gather, 0 FLOPs -> bandwidth problem (~7 us roofline at 23.3 TB/s with
// the 40 MB table L2-resident). CDNA5 path: async global->LDS->global B128
// copies tracked by ASYNCcnt; data never round-trips through VGPRs.

#define AS1 __attribute__((address_space(1)))
#define AS3 __attribute__((address_space(3)))

typedef float v4f  __attribute__((ext_vector_type(4)));
typedef int   b128_t __attribute__((vector_size(16)));   // matches builtin param

#if defined(__AMDGCN__) &&                                                  \
    __has_builtin(__builtin_amdgcn_global_load_async_to_lds_b128) &&        \
    __has_builtin(__builtin_amdgcn_global_store_async_from_lds_b128) &&     \
    __has_builtin(__builtin_amdgcn_s_wait_asynccnt)
#define USE_ASYNC_LDS 1
#elif defined(__AMDGCN__)
#warning "gfx1250 async-to-LDS builtins not found on this toolchain; falling back to direct VGPR B128 copy path"
#define USE_ASYNC_LDS 0
#else
#define USE_ASYNC_LDS 0
#endif

__global__ __launch_bounds__(256) void hash_embed_gather(
    const int* __restrict__ idx,     // [nrows] bucket ids (0..9999)
    const v4f* __restrict__ emb,     // [10000 * 256] float4 rows
    v4f* __restrict__ out,           // [nrows * 256] float4
    int nrows)
{
#if USE_ASYNC_LDS
  // One wave per row: 8 async B128 loads (32 lanes * 16 B = 512 B each) into
  // a private 4 KiB LDS slice, drain ASYNCcnt, then 8 async B128 stores to
  // the contiguous output row. MLP comes from 32768 concurrent waves.
  __shared__ char lbuf[8 * 4096];                 // 4 KiB per wave, 8 waves
  const int lane = threadIdx.x & 31;
  const int wib  = threadIdx.x >> 5;
  char* mybuf = &lbuf[wib * 4096 + lane * 16];

  const int gw = blockIdx.x * 8 + wib;            // global wave id
  const int nw = gridDim.x * 8;

  for (int r = gw; r < nrows; r += nw) {
    const int row = __builtin_amdgcn_readfirstlane(idx[r]);  // wave-uniform
    char* src = (char*)emb + (size_t)row * 4096 + (size_t)(lane * 16);
    char* dst = (char*)out + (size_t)r   * 4096 + (size_t)(lane * 16);

#pragma unroll
    for (int s = 0; s < 8; ++s)
      __builtin_amdgcn_global_load_async_to_lds_b128(
          (AS1 b128_t*)(src + s * 512), (AS3 b128_t*)(mybuf + s * 512),
          /*offset=*/0, /*cpol=*/0);

    __builtin_amdgcn_s_wait_asynccnt(0);          // loads landed in LDS

#pragma unroll
    for (int s = 0; s < 8; ++s)
      __builtin_amdgcn_global_store_async_from_lds_b128(
          (AS1 b128_t*)(dst + s * 512), (AS3 b128_t*)(mybuf + s * 512),
          /*offset=*/0, /*cpol=*/0);

    __builtin_amdgcn_s_wait_asynccnt(0);          // LDS slice reusable
  }
#else
  // Direct path: 4 rows per block iteration. Each thread moves one float4
  // per row (256 threads * 16 B = 4 KiB = one row). Index loads are
  // block-uniform -> scalar. Output is write-once: non-temporal stores keep
  // the 40 MB embedding table resident in the 192 MB L2.
  const int tid = threadIdx.x;
  for (int r0 = blockIdx.x * 4; r0 < nrows; r0 += gridDim.x * 4) {
#pragma unroll
    for (int k = 0; k < 4; ++k) {
      const int r = r0 + k;                       // nrows % 4 == 0
      const int row = idx[r];                     // uniform -> s_load
      v4f v = emb[(size_t)row * 256 + tid];       // global_load_b128
      __builtin_nontemporal_store(v, &out[(size_t)r * 256 + tid]);
    }
  }
#endif
}

extern "C" void kernel_launch(void* const* d_in, const int* in_sizes, int n_in,
                              void* d_out, int out_size, void* d_ws, size_t ws_size,
                              hipStream_t stream) {
  // setup_inputs order: 0=input_ids (unused), 1=hashed_ids, 2=embedding
  const int*  hashed = (const int*)d_in[1];
  const v4f*  emb    = (const v4f*)d_in[2];
  v4f*        out    = (v4f*)d_out;
  const int   nrows  = in_sizes[1];               // 32768

  // 4096 blocks x 256 threads:
  //  - async path: 32768 waves -> exactly one 4 KiB row per wave
  //  - direct path: 4 rows per block per grid-stride iteration
  hash_embed_gather<<<dim3(4096), dim3(256), 0, stream>>>(hashed, emb, out, nrows);
}